// GeneInteractionGraph_81389630259484
// MI455X (gfx1250) — compile-verified
//
#include <hip/hip_runtime.h>
#include <hip/hip_bf16.h>

typedef __attribute__((ext_vector_type(2))) float v2f;
typedef __attribute__((ext_vector_type(8))) float v8f;

#define FEAT 256
#define A_STRIDE 260   // pad: 260%64=4 -> conflict-free frag reads; 260*4%16==0 -> float4 staging aligned

// ---------------- degree / CSR construction ----------------

__global__ __launch_bounds__(256) void zero_counts_kernel(int* counts, int n) {
    int i = blockIdx.x * blockDim.x + threadIdx.x;
    if (i < n) counts[i] = 0;
}

__global__ __launch_bounds__(256) void count_kernel(const int* __restrict__ dst, int* counts, int E) {
    int e = blockIdx.x * blockDim.x + threadIdx.x;
    if (e < E) atomicAdd(&counts[dst[e]], 1);
}

// Single-block exclusive scan over counts -> row_start; also dinv = rsqrt(deg+1); reset counts (cursor).
__global__ __launch_bounds__(1024) void scan_kernel(int* counts, int* row_start, float* dinv, int n, int E) {
    __shared__ int sdata[1024];
    __shared__ int s_carry;
    if (threadIdx.x == 0) s_carry = 0;
    __syncthreads();
    for (int base = 0; base < n; base += 1024) {
        int i = base + (int)threadIdx.x;
        int orig = (i < n) ? counts[i] : 0;
        sdata[threadIdx.x] = orig;
        __syncthreads();
        // Hillis-Steele inclusive scan
        for (int off = 1; off < 1024; off <<= 1) {
            int t = (threadIdx.x >= (unsigned)off) ? sdata[threadIdx.x - off] : 0;
            __syncthreads();
            sdata[threadIdx.x] += t;
            __syncthreads();
        }
        int incl = sdata[threadIdx.x];
        int excl = incl - orig;
        if (i < n) {
            row_start[i] = s_carry + excl;
            dinv[i] = rsqrtf((float)orig + 1.0f);
            counts[i] = 0;   // becomes cursor for fill pass
        }
        __syncthreads();
        if (threadIdx.x == 1023) s_carry += incl;
        __syncthreads();
    }
    if (threadIdx.x == 0) row_start[n] = E;
}

__global__ __launch_bounds__(256) void fill_kernel(const int* __restrict__ src, const int* __restrict__ dst,
                                                   const int* __restrict__ row_start, int* cursor,
                                                   int* __restrict__ csr_src, int E) {
    int e = blockIdx.x * blockDim.x + threadIdx.x;
    if (e < E) {
        int d = dst[e];
        int pos = atomicAdd(&cursor[d], 1);
        csr_src[row_start[d] + pos] = src[e];
    }
}

// ---------------- dense GEMM: H = X @ W (fp32 WMMA 16x16x4) ----------------
// grid = (2, rows/16); block = 256 (8 waves); wave w handles col-tile blockIdx.x*8 + w.
__global__ __launch_bounds__(256) void gemm_kernel(const float* __restrict__ X, const float* __restrict__ W,
                                                   float* __restrict__ H) {
    __shared__ float As[16 * A_STRIDE];
    const int rowBase = blockIdx.y * 16;
    const int waveId  = threadIdx.x >> 5;
    const int lane    = threadIdx.x & 31;
    const int colBase = (blockIdx.x * 8 + waveId) * 16;

    // Stage the 16x256 A strip into LDS (coalesced float4 loads, 16B-aligned stores).
    for (int idx = threadIdx.x; idx < 16 * 64; idx += 256) {
        int r  = idx >> 6;
        int c4 = (idx & 63) << 2;
        float4 v = *(const float4*)(X + (size_t)(rowBase + r) * FEAT + c4);
        *(float4*)&As[r * A_STRIDE + c4] = v;
    }
    __syncthreads();

    const int half = lane >> 4;    // 0: K+0/K+1 ; 1: K+2/K+3
    const int l15  = lane & 15;

    v8f c = {0.f, 0.f, 0.f, 0.f, 0.f, 0.f, 0.f, 0.f};
    for (int k = 0; k < FEAT; k += 4) {
        // A frag (16x4): lane row = l15; VGPR0/1 = cols k+half*2, k+half*2+1  -> contiguous b64 LDS read
        v2f a = *(const v2f*)&As[l15 * A_STRIDE + k + half * 2];
        // B frag (4x16): lane col = l15; VGPR0/1 = rows k+half*2, k+half*2+1
        const float* wp = W + (size_t)(k + half * 2) * FEAT + colBase + l15;
        v2f b;
        b.x = wp[0];
        b.y = wp[FEAT];
        c = __builtin_amdgcn_wmma_f32_16x16x4_f32(false, a, false, b, (short)0, c, false, false);
    }

    // D layout: VGPR j, lanes 0-15 -> row j, lanes 16-31 -> row j+8; col = l15
    float* out = H + (size_t)rowBase * FEAT + colBase + l15;
#pragma unroll
    for (int j = 0; j < 8; ++j) {
        out[(size_t)(j + half * 8) * FEAT] = c[j];
    }
}

// ---------------- per-node aggregation: out = agg + h*dinv^2 + b (opt. ReLU) ----------------
// One wave (32 lanes) per node; lane l owns features [4l,4l+4) and [128+4l,128+4l+4) as float4s.
// Per edge: two global_load_b128 (512 B each across the wave) from an SGPR row base.

__device__ __forceinline__ float4 fma4(float4 a, float s, float4 acc) {
    acc.x = fmaf(a.x, s, acc.x);
    acc.y = fmaf(a.y, s, acc.y);
    acc.z = fmaf(a.z, s, acc.z);
    acc.w = fmaf(a.w, s, acc.w);
    return acc;
}

__global__ __launch_bounds__(256) void agg_kernel(const float* __restrict__ h, const float* __restrict__ dinv,
                                                  const int* __restrict__ row_start, const int* __restrict__ csr_src,
                                                  const float* __restrict__ bias, float* __restrict__ out,
                                                  int do_relu, int n) {
    const int wave = threadIdx.x >> 5;
    const int lane = threadIdx.x & 31;
    const int node = blockIdx.x * 8 + wave;
    if (node >= n) return;

    const float di = dinv[node];
    const int s0 = row_start[node];
    const int s1 = row_start[node + 1];

    const float4* __restrict__ h4 = (const float4*)h;   // 64 float4 per row

    float4 acc0 = {0.f, 0.f, 0.f, 0.f};
    float4 acc1 = {0.f, 0.f, 0.f, 0.f};

    for (int p = s0; p < s1; ++p) {
        // wave-uniform: force into SGPRs -> saddr-form loads, scalar dinv fetch
        int s = __builtin_amdgcn_readfirstlane(csr_src[p]);
        float nrm = dinv[s] * di;
        const float4* row = h4 + (size_t)s * 64;
        float4 a = row[lane];
        float4 b = row[32 + lane];
        acc0 = fma4(a, nrm, acc0);
        acc1 = fma4(b, nrm, acc1);
    }

    const float sc = di * di;
    const float4* srow = h4 + (size_t)node * 64;
    float4 hs0 = srow[lane];
    float4 hs1 = srow[32 + lane];
    const float4* b4 = (const float4*)bias;
    float4 bb0 = b4[lane];
    float4 bb1 = b4[32 + lane];

    float4 r0, r1;
    r0.x = acc0.x + hs0.x * sc + bb0.x;
    r0.y = acc0.y + hs0.y * sc + bb0.y;
    r0.z = acc0.z + hs0.z * sc + bb0.z;
    r0.w = acc0.w + hs0.w * sc + bb0.w;
    r1.x = acc1.x + hs1.x * sc + bb1.x;
    r1.y = acc1.y + hs1.y * sc + bb1.y;
    r1.z = acc1.z + hs1.z * sc + bb1.z;
    r1.w = acc1.w + hs1.w * sc + bb1.w;

    if (do_relu) {
        r0.x = fmaxf(r0.x, 0.f); r0.y = fmaxf(r0.y, 0.f); r0.z = fmaxf(r0.z, 0.f); r0.w = fmaxf(r0.w, 0.f);
        r1.x = fmaxf(r1.x, 0.f); r1.y = fmaxf(r1.y, 0.f); r1.z = fmaxf(r1.z, 0.f); r1.w = fmaxf(r1.w, 0.f);
    }

    float4* o4 = (float4*)out + (size_t)node * 64;
    o4[lane]      = r0;
    o4[32 + lane] = r1;
}

// ---------------- launcher ----------------

static inline size_t align256(size_t x) { return (x + 255) & ~(size_t)255; }

extern "C" void kernel_launch(void* const* d_in, const int* in_sizes, int n_in,
                              void* d_out, int out_size, void* d_ws, size_t ws_size,
                              hipStream_t stream) {
    // inputs: 0 gene_ind_vec (unused), 1 edge_index (2,2*Npairs) int32, 2 embedding (N,256) f32,
    //         3 W1 (256,256), 4 b1 (256), 5 W2 (256,256), 6 b2 (256)
    const int*   edge = (const int*)d_in[1];
    const float* X    = (const float*)d_in[2];
    const float* W1   = (const float*)d_in[3];
    const float* b1   = (const float*)d_in[4];
    const float* W2   = (const float*)d_in[5];
    const float* b2   = (const float*)d_in[6];
    float* out = (float*)d_out;

    const int E = in_sizes[1] / 2;        // directed edges (1,000,000)
    const int N = in_sizes[2] / FEAT;     // nodes (50,000)
    const int* src = edge;
    const int* dst = edge + E;

    // workspace layout
    char* ws = (char*)d_ws;
    size_t off = 0;
    int*   counts    = (int*)(ws + off);  off += align256((size_t)N * 4);
    int*   row_start = (int*)(ws + off);  off += align256((size_t)(N + 1) * 4);
    float* dinv      = (float*)(ws + off); off += align256((size_t)N * 4);
    int*   csr_src   = (int*)(ws + off);  off += align256((size_t)E * 4);
    float* bufA      = (float*)(ws + off); off += align256((size_t)N * FEAT * 4);
    float* bufB      = (float*)(ws + off); off += align256((size_t)N * FEAT * 4);
    (void)ws_size; (void)n_in; (void)out_size;

    const int TB = 256;
    dim3 gemm_grid(2, N / 16);
    const int agg_blocks = (N + 7) / 8;   // 8 waves per block, one node per wave

    // --- build normalization + CSR ---
    zero_counts_kernel<<<(N + TB - 1) / TB, TB, 0, stream>>>(counts, N);
    count_kernel<<<(E + TB - 1) / TB, TB, 0, stream>>>(dst, counts, E);
    scan_kernel<<<1, 1024, 0, stream>>>(counts, row_start, dinv, N, E);
    fill_kernel<<<(E + TB - 1) / TB, TB, 0, stream>>>(src, dst, row_start, counts, csr_src, E);

    // --- layer 1: h1 = X @ W1 ; x1 = relu(agg(h1) + h1*dinv^2 + b1) ---
    gemm_kernel<<<gemm_grid, TB, 0, stream>>>(X, W1, bufA);
    agg_kernel<<<agg_blocks, TB, 0, stream>>>(bufA, dinv, row_start, csr_src, b1, bufB, 1, N);

    // --- layer 2: h2 = x1 @ W2 ; out = agg(h2) + h2*dinv^2 + b2 ---
    gemm_kernel<<<gemm_grid, TB, 0, stream>>>(bufB, W2, bufA);
    agg_kernel<<<agg_blocks, TB, 0, stream>>>(bufA, dinv, row_start, csr_src, b2, out, 0, N);
}